// GIoULoss_3788161155127
// MI455X (gfx1250) — compile-verified
//
#include <hip/hip_runtime.h>
#include <hip/hip_bf16.h>

#define TPB    256      // 8 wave32 waves per workgroup on CDNA5
#define BLOCKS 2048     // 512K threads -> ~15 boxes/thread grid-stride

// Native clang vector type: accepted by __builtin_nontemporal_load and lowers
// to a single global_load_b128 th:NT (boxes are 16-byte aligned float4s).
typedef float v4f __attribute__((ext_vector_type(4)));

// Branchless _fix_boxes: if x1>x2 or y1>y2, clamp all coords to min=0.
// Per-component selects lower to v_cmp + v_cndmask (no exec-mask branching).
__device__ __forceinline__ v4f fix_box(v4f b)
{
    const bool bad = (b.x > b.z) | (b.y > b.w);
    v4f r;
    r.x = bad ? fmaxf(b.x, 0.0f) : b.x;
    r.y = bad ? fmaxf(b.y, 0.0f) : b.y;
    r.z = bad ? fmaxf(b.z, 0.0f) : b.z;
    r.w = bad ? fmaxf(b.w, 0.0f) : b.w;
    return r;
}

// clip(1 - GIoU(p, t), 0, inf) for one box pair.
__device__ __forceinline__ float giou_loss(v4f p, v4f t)
{
    p = fix_box(p);
    t = fix_box(t);

    const float area_p = (p.z - p.x) * (p.w - p.y);
    const float area_t = (t.z - t.x) * (t.w - t.y);

    // intersection
    const float iw = fmaxf(fminf(p.z, t.z) - fmaxf(p.x, t.x), 0.0f);
    const float ih = fmaxf(fminf(p.w, t.w) - fmaxf(p.y, t.y), 0.0f);
    const float inter = iw * ih;

    const float uni = area_p + area_t - inter;
    const float iou = inter / uni;

    // smallest enclosing box
    const float cw = fmaxf(fmaxf(p.z, t.z) - fminf(p.x, t.x), 0.0f);
    const float ch = fmaxf(fmaxf(p.w, t.w) - fminf(p.y, t.y), 0.0f);
    const float area_c = cw * ch;

    const float giou = iou - (area_c - uni) / area_c;
    return fmaxf(1.0f - giou, 0.0f);
}

// ---------------------------------------------------------------------------
// Pass 1: per-block partial sums. Streaming NT b128 loads (256 MB read once
// > 192 MB L2), 2x unrolled grid-stride for 4 outstanding b128 loads/thread,
// wave32 shuffle tree + 8-wave LDS combine.
// ---------------------------------------------------------------------------
__global__ __launch_bounds__(TPB)
void giou_partial_kernel(const v4f* __restrict__ pred,
                         const v4f* __restrict__ targ,
                         float* __restrict__ partial,
                         int n)
{
    const int tid    = blockIdx.x * blockDim.x + threadIdx.x;
    const int stride = gridDim.x * blockDim.x;

    float acc0 = 0.0f, acc1 = 0.0f;

    int i = tid;
    for (; i + stride < n; i += 2 * stride) {
        // Issue all four b128 loads before the dependent ALU chains.
        const v4f p0 = __builtin_nontemporal_load(&pred[i]);
        const v4f t0 = __builtin_nontemporal_load(&targ[i]);
        const v4f p1 = __builtin_nontemporal_load(&pred[i + stride]);
        const v4f t1 = __builtin_nontemporal_load(&targ[i + stride]);
        acc0 += giou_loss(p0, t0);
        acc1 += giou_loss(p1, t1);
    }
    for (; i < n; i += stride) {
        const v4f p = __builtin_nontemporal_load(&pred[i]);
        const v4f t = __builtin_nontemporal_load(&targ[i]);
        acc0 += giou_loss(p, t);
    }

    float acc = acc0 + acc1;

    // ---- wave32 reduction (5 xor-shuffle steps; warpSize == 32 on gfx1250)
    #pragma unroll
    for (int off = 16; off > 0; off >>= 1)
        acc += __shfl_xor(acc, off, 32);

    // ---- cross-wave combine: 8 waves per block
    __shared__ float lds[TPB / 32];
    const int lane = threadIdx.x & 31;
    const int wave = threadIdx.x >> 5;
    if (lane == 0) lds[wave] = acc;
    __syncthreads();

    if (wave == 0) {
        float v = (lane < (TPB / 32)) ? lds[lane] : 0.0f;
        #pragma unroll
        for (int off = 4; off > 0; off >>= 1)   // reduce 8 values (stays in group-of-8)
            v += __shfl_xor(v, off, 32);
        if (lane == 0) partial[blockIdx.x] = v;
    }
}

// ---------------------------------------------------------------------------
// Pass 2: tree-reduce the block partials and scale by 1/N. Single block,
// fully deterministic (no atomics), low rounding error via tree structure.
// ---------------------------------------------------------------------------
__global__ __launch_bounds__(TPB)
void giou_final_kernel(const float* __restrict__ partial,
                       float* __restrict__ out,
                       int nblocks, float inv_n)
{
    float acc = 0.0f;
    for (int i = threadIdx.x; i < nblocks; i += TPB)
        acc += partial[i];

    #pragma unroll
    for (int off = 16; off > 0; off >>= 1)
        acc += __shfl_xor(acc, off, 32);

    __shared__ float lds[TPB / 32];
    const int lane = threadIdx.x & 31;
    const int wave = threadIdx.x >> 5;
    if (lane == 0) lds[wave] = acc;
    __syncthreads();

    if (wave == 0) {
        float v = (lane < (TPB / 32)) ? lds[lane] : 0.0f;
        #pragma unroll
        for (int off = 4; off > 0; off >>= 1)
            v += __shfl_xor(v, off, 32);
        if (lane == 0) out[0] = v * inv_n;
    }
}

// ---------------------------------------------------------------------------
// Host-side launcher (graph-capture safe: only kernel launches on stream).
// ---------------------------------------------------------------------------
extern "C" void kernel_launch(void* const* d_in, const int* in_sizes, int n_in,
                              void* d_out, int out_size, void* d_ws, size_t ws_size,
                              hipStream_t stream)
{
    const v4f* pred = (const v4f*)d_in[0];   // [N,4] float32
    const v4f* targ = (const v4f*)d_in[1];   // [N,4] float32
    float*     out  = (float*)d_out;         // scalar float32
    float*     part = (float*)d_ws;          // BLOCKS partial sums

    const int n = in_sizes[0] / 4;           // number of boxes

    int nblocks = BLOCKS;
    if ((size_t)nblocks * sizeof(float) > ws_size)
        nblocks = (int)(ws_size / sizeof(float));  // defensive; ws is plenty in practice

    giou_partial_kernel<<<nblocks, TPB, 0, stream>>>(pred, targ, part, n);
    giou_final_kernel<<<1, TPB, 0, stream>>>(part, out, nblocks, 1.0f / (float)n);
}